// SCRFDWithPost_23691039605072
// MI455X (gfx1250) — compile-verified
//
#include <hip/hip_runtime.h>
#include <math.h>

// ---------------- problem constants (from reference) ----------------
#define N8   130560   // (2176/8)*(3840/8)  = 272*480
#define N16  32640    // 136*240
#define N32  8160     // 68*120
#define NTOT (N8 + N16 + N32)
#define TOPK 1000
#define PAD  1024
#define NMS_T 0.4f
#define CAND_MAX 4096

// ---------------- workspace layout (bytes) ----------------
#define HIST_OFF   0           // 65536 * 4 = 262144
#define CNT_OFF    262144      // 64 bytes: [0]=cand_count, [1]=threshold bin
#define CAND_OFF   262208      // u64[4096] = 32768
#define SELIDX_OFF 294976      // u32[1024]
#define SELSC_OFF  299072      // f32[1024]
#define BOXES_OFF  303168      // f32[1024*4] (16B aligned)
#define AREA_OFF   319552      // f32[1024]
#define KEEP_OFF   323648      // u32[1024]
#define KPS_OFF    327744      // f32[1024*10]
#define IOU_OFF    368704      // f32[1024*1024] = 4,194,304 -> total ~4.56 MB

typedef __attribute__((ext_vector_type(2))) float v2f;
typedef __attribute__((ext_vector_type(8))) float v8f;

// ---------------- kernels ----------------

__global__ void zero_u32_kernel(unsigned* __restrict__ p, int n) {
  int i = blockIdx.x * blockDim.x + threadIdx.x;
  if (i < n) p[i] = 0u;
}

__device__ __forceinline__ unsigned score_bin(unsigned bits) {
  // s in (0.5, 1.0] -> bits in (0x3F000000, 0x3F800000]; 16-bit monotone bin
  unsigned b = (bits - 0x3F000000u) >> 7;
  return (b > 65535u) ? 65535u : b;
}

// Pass 1: histogram of candidate score bits (only scores streamed: ~685 KB total)
__global__ void hist_kernel(const float* __restrict__ sc, int n,
                            unsigned* __restrict__ hist) {
  int i = blockIdx.x * blockDim.x + threadIdx.x;
  if (i >= n) return;
  __builtin_prefetch(sc + i + 16384, 0, 1);   // global_prefetch_b8
  float x = sc[i];
  float s = 1.0f / (1.0f + __expf(-x));
  if (s > 0.5f) {
    atomicAdd(&hist[score_bin(__float_as_uint(s))], 1u);
  }
}

// Pass 2: radix-select threshold bin T = max b with suffix_count(b) >= TOPK
__global__ void __launch_bounds__(1024) thresh_kernel(const unsigned* __restrict__ hist,
                                                      unsigned* __restrict__ cnt) {
  __shared__ unsigned suf[1024];
  int t = threadIdx.x;
  unsigned s = 0;
  for (int b = 0; b < 64; ++b) s += hist[t * 64 + b];
  suf[t] = s;
  __syncthreads();
  for (int off = 1; off < 1024; off <<= 1) {
    unsigned v = (t + off < 1024) ? suf[t + off] : 0u;
    __syncthreads();
    suf[t] += v;
    __syncthreads();
  }
  unsigned myS = suf[t];
  unsigned nxS = (t + 1 < 1024) ? suf[t + 1] : 0u;
  if (myS >= TOPK && nxS < TOPK) {
    unsigned cum = nxS;
    int T = t * 64;
    for (int b = t * 64 + 63; b >= t * 64; --b) {
      cum += hist[b];
      if (cum >= TOPK) { T = b; break; }
    }
    cnt[1] = (unsigned)T;
  }
  if (t == 0 && suf[0] < TOPK) cnt[1] = 0u;  // fewer than TOPK candidates: take all
}

// Pass 3: compact candidates in bins >= T as 64-bit sort keys (score bits | ~idx)
__global__ void compact_kernel(const float* __restrict__ sc, int n, int base,
                               unsigned* __restrict__ cnt,
                               unsigned long long* __restrict__ cand) {
  int i = blockIdx.x * blockDim.x + threadIdx.x;
  if (i >= n) return;
  float x = sc[i];
  float s = 1.0f / (1.0f + __expf(-x));
  if (s > 0.5f) {
    unsigned bits = __float_as_uint(s);
    if (score_bin(bits) >= cnt[1]) {
      unsigned pos = atomicAdd(&cnt[0], 1u);
      if (pos < CAND_MAX) {
        unsigned idx = (unsigned)(base + i);
        cand[pos] = ((unsigned long long)bits << 32) |
                    (unsigned long long)(~idx);       // ties: smaller idx first
      }
    }
  }
}

// Pass 4: bitonic sort (descending) of <=4096 survivors in LDS; emit sorted top-1000
__global__ void __launch_bounds__(1024) sort_kernel(const unsigned* __restrict__ cnt,
                                                    const unsigned long long* __restrict__ cand,
                                                    unsigned* __restrict__ selIdx,
                                                    float* __restrict__ selSc,
                                                    unsigned* __restrict__ valid) {
  __shared__ unsigned long long key[CAND_MAX];
  int t = threadIdx.x;
  unsigned count = cnt[0];
  if (count > CAND_MAX) count = CAND_MAX;
  for (int i = t; i < CAND_MAX; i += 1024)
    key[i] = (i < (int)count) ? cand[i] : 0ull;
  __syncthreads();
  for (int k = 2; k <= CAND_MAX; k <<= 1) {
    for (int j = k >> 1; j > 0; j >>= 1) {
      for (int i = t; i < CAND_MAX; i += 1024) {
        int l = i ^ j;
        if (l > i) {
          bool up = ((i & k) == 0);
          unsigned long long a = key[i], b = key[l];
          if (up ? (a < b) : (a > b)) { key[i] = b; key[l] = a; }
        }
      }
      __syncthreads();
    }
  }
  if (t < TOPK) {
    unsigned long long kk = key[t];
    if (kk != 0ull) {
      selIdx[t] = ~(unsigned)(kk & 0xFFFFFFFFull);
      selSc[t]  = __uint_as_float((unsigned)(kk >> 32));
      valid[t]  = 1u;
    } else {
      selIdx[t] = 0u; selSc[t] = 0.0f; valid[t] = 0u;
    }
  }
}

// Pass 5: decode boxes/keypoints only for the 1000 winners (tiny gather)
__global__ void __launch_bounds__(1024) decode_kernel(
    const unsigned* __restrict__ selIdx, const unsigned* __restrict__ valid,
    const float* __restrict__ bb8,  const float* __restrict__ kp8,
    const float* __restrict__ bb16, const float* __restrict__ kp16,
    const float* __restrict__ bb32, const float* __restrict__ kp32,
    float* __restrict__ boxes, float* __restrict__ area, float* __restrict__ kps) {
  int t = threadIdx.x;
  if (t >= TOPK || !valid[t]) return;
  unsigned idx = selIdx[t];
  const float* bb; const float* kp; int li; float strd; int fw;
  if (idx < N8)            { bb = bb8;  kp = kp8;  li = (int)idx;             strd = 8.0f;  fw = 480; }
  else if (idx < N8 + N16) { bb = bb16; kp = kp16; li = (int)idx - N8;        strd = 16.0f; fw = 240; }
  else                     { bb = bb32; kp = kp32; li = (int)idx - N8 - N16;  strd = 32.0f; fw = 120; }
  float cx = (float)(li % fw) * strd;
  float cy = (float)(li / fw) * strd;
  float x1 = cx - bb[li * 4 + 0] * strd;
  float y1 = cy - bb[li * 4 + 1] * strd;
  float x2 = cx + bb[li * 4 + 2] * strd;
  float y2 = cy + bb[li * 4 + 3] * strd;
  boxes[t * 4 + 0] = x1; boxes[t * 4 + 1] = y1;
  boxes[t * 4 + 2] = x2; boxes[t * 4 + 3] = y2;
  area[t] = (x2 - x1) * (y2 - y1);
  for (int p = 0; p < 5; ++p) {
    kps[t * 10 + 2 * p + 0] = kp[li * 10 + 2 * p + 0] * strd + cx;
    kps[t * 10 + 2 * p + 1] = kp[li * 10 + 2 * p + 1] * strd + cy;
  }
}

// Pass 6: 16x16 IoU tiles. One wave32 per tile, fully straight-line:
//  - area_i + area_j via a single v_wmma_f32_16x16x4_f32 (A=[area_i,1] x B=[1;area_j])
//  - boxes staged into LDS via async tensor path (global_load_async_to_lds_b128)
//  - iou matrix padded to 1024x1024 so every lane/row store is unconditional
__global__ void __launch_bounds__(32) iou_kernel(const float* __restrict__ boxes,
                                                 const float* __restrict__ area,
                                                 float* __restrict__ iou) {
  __shared__ float sbox[32][4];  // [0..15] = I-tile rows, [16..31] = J-tile rows
  int lane = threadIdx.x;
  int I0 = blockIdx.x * 16, J0 = blockIdx.y * 16;
  int m = lane & 15;
  bool hiHalf = lane >= 16;

  // async stage: lane l<16 loads box[I0+l]; lane l>=16 loads box[J0+l-16]
  unsigned ldsAddr = (unsigned)(unsigned long long)(&sbox[lane][0]);
  unsigned row = hiHalf ? (unsigned)(J0 + m) : (unsigned)(I0 + m);
  unsigned goff = row * 16u;
  asm volatile("global_load_async_to_lds_b128 %0, %1, %2"
               :: "v"(ldsAddr), "v"(goff), "s"(boxes) : "memory");

  // WMMA overlapped with the async DMA. Unconditional area loads (rows < 1024
  // always in-bounds) + cndmask selects keep EXEC all-1s through the WMMA.
  // A (16x4): VGPR0 = K0 (lanes0-15) | K2 (lanes16-31); VGPR1 = K1 | K3.
  // B (4x16): VGPR0 = K0-row | K2-row ; VGPR1 = K1-row | K3-row.
  float aI = area[I0 + m];
  float aJ = area[J0 + m];
  v2f a, b;
  a.x = hiHalf ? 0.0f : aI;    // A[m][0] = area_i ; A[m][2] = 0
  a.y = hiHalf ? 0.0f : 1.0f;  // A[m][1] = 1      ; A[m][3] = 0
  b.x = hiHalf ? 0.0f : 1.0f;  // B[0][n] = 1      ; B[2][n] = 0
  b.y = hiHalf ? 0.0f : aJ;    // B[1][n] = area_j ; B[3][n] = 0
  v8f c = {};
  c = __builtin_amdgcn_wmma_f32_16x16x4_f32(false, a, false, b,
                                            (short)0, c, false, false);

  asm volatile("s_wait_asynccnt 0" ::: "memory");

  int gj = J0 + m;
  float jx1 = sbox[16 + m][0], jy1 = sbox[16 + m][1];
  float jx2 = sbox[16 + m][2], jy2 = sbox[16 + m][3];
  int mrow = hiHalf ? 8 : 0;
#pragma unroll
  for (int r = 0; r < 8; ++r) {
    int M = r + mrow;               // D layout: vgpr r holds rows M=r and M=r+8
    float ix1 = sbox[M][0], iy1 = sbox[M][1];
    float ix2 = sbox[M][2], iy2 = sbox[M][3];
    float lx = fmaxf(ix1, jx1), ly = fmaxf(iy1, jy1);
    float rx = fminf(ix2, jx2), ry = fminf(iy2, jy2);
    float inter = fmaxf(rx - lx, 0.0f) * fmaxf(ry - ly, 0.0f);
    float denom = fmaxf(c[r] - inter, 1e-9f);   // c[r] = area_i + area_j (WMMA)
    // iou only feeds the internal > NMS_T compare -> hw reciprocal is fine
    iou[(I0 + M) * PAD + gj] = inter * __builtin_amdgcn_rcpf(denom);
  }
}

// Pass 7: greedy NMS (exact recurrence of reference fori_loop) + masked output
__global__ void __launch_bounds__(1024) nms_out_kernel(
    const unsigned* __restrict__ validIn, const float* __restrict__ iou,
    const float* __restrict__ selSc, const float* __restrict__ boxes,
    const float* __restrict__ kps, float* __restrict__ out) {
  __shared__ unsigned keep[TOPK];
  int t = threadIdx.x;
  if (t < TOPK) keep[t] = validIn[t];
  __syncthreads();
  for (int i = 0; i < TOPK; ++i) {
    unsigned ki = keep[i];
    if (ki && t > i && t < TOPK && iou[i * PAD + t] > NMS_T) keep[t] = 0u;
    __syncthreads();
  }
  if (t < TOPK) {
    unsigned k = keep[t];
    out[4000 + t] = k ? selSc[t] : 0.0f;                       // final_scores
    for (int cI = 0; cI < 4; ++cI)
      out[t * 4 + cI] = k ? boxes[t * 4 + cI] : 0.0f;          // final_boxes
    for (int p = 0; p < 10; ++p)
      out[5000 + t * 10 + p] = k ? kps[t * 10 + p] : 0.0f;     // final_kpss
  }
}

// ---------------- launch ----------------
extern "C" void kernel_launch(void* const* d_in, const int* in_sizes, int n_in,
                              void* d_out, int out_size, void* d_ws, size_t ws_size,
                              hipStream_t stream) {
  // setup_inputs order: x, scores8, bbox8, kps8, scores16, bbox16, kps16,
  //                     scores32, bbox32, kps32   (x unused by reference)
  const float* sc8  = (const float*)d_in[1];
  const float* bb8  = (const float*)d_in[2];
  const float* kp8  = (const float*)d_in[3];
  const float* sc16 = (const float*)d_in[4];
  const float* bb16 = (const float*)d_in[5];
  const float* kp16 = (const float*)d_in[6];
  const float* sc32 = (const float*)d_in[7];
  const float* bb32 = (const float*)d_in[8];
  const float* kp32 = (const float*)d_in[9];
  float* out = (float*)d_out;

  char* ws = (char*)d_ws;
  unsigned*           hist   = (unsigned*)(ws + HIST_OFF);
  unsigned*           cnt    = (unsigned*)(ws + CNT_OFF);
  unsigned long long* cand   = (unsigned long long*)(ws + CAND_OFF);
  unsigned*           selIdx = (unsigned*)(ws + SELIDX_OFF);
  float*              selSc  = (float*)(ws + SELSC_OFF);
  float*              boxes  = (float*)(ws + BOXES_OFF);
  float*              area   = (float*)(ws + AREA_OFF);
  unsigned*           keepV  = (unsigned*)(ws + KEEP_OFF);
  float*              kps    = (float*)(ws + KPS_OFF);
  float*              iou    = (float*)(ws + IOU_OFF);

  // zero histogram + counters (contiguous), and boxes/area/keep/kps block
  zero_u32_kernel<<<(65552 + 1023) / 1024, 1024, 0, stream>>>(hist, 65552);
  zero_u32_kernel<<<(16384 + 1023) / 1024, 1024, 0, stream>>>(
      (unsigned*)(ws + BOXES_OFF), 16384);  // 65536 B covering boxes..kps

  // pass 1: histogram (scores only, ~685 KB streamed)
  hist_kernel<<<(N8  + 255) / 256, 256, 0, stream>>>(sc8,  N8,  hist);
  hist_kernel<<<(N16 + 255) / 256, 256, 0, stream>>>(sc16, N16, hist);
  hist_kernel<<<(N32 + 255) / 256, 256, 0, stream>>>(sc32, N32, hist);

  // pass 2: select threshold bin
  thresh_kernel<<<1, 1024, 0, stream>>>(hist, cnt);

  // pass 3: compact survivors
  compact_kernel<<<(N8  + 255) / 256, 256, 0, stream>>>(sc8,  N8,  0,        cnt, cand);
  compact_kernel<<<(N16 + 255) / 256, 256, 0, stream>>>(sc16, N16, N8,       cnt, cand);
  compact_kernel<<<(N32 + 255) / 256, 256, 0, stream>>>(sc32, N32, N8 + N16, cnt, cand);

  // pass 4: sorted top-1000
  sort_kernel<<<1, 1024, 0, stream>>>(cnt, cand, selIdx, selSc, keepV);

  // pass 5: decode winners only
  decode_kernel<<<1, 1024, 0, stream>>>(selIdx, keepV, bb8, kp8, bb16, kp16,
                                        bb32, kp32, boxes, area, kps);

  // pass 6: IoU matrix, one wave32 per 16x16 tile (WMMA + async-LDS staging)
  dim3 tg((TOPK + 15) / 16, (TOPK + 15) / 16);
  iou_kernel<<<tg, 32, 0, stream>>>(boxes, area, iou);

  // pass 7: greedy NMS + masked outputs
  nms_out_kernel<<<1, 1024, 0, stream>>>(keepV, iou, selSc, boxes, kps, out);
}